// RGCN_39462159515659
// MI455X (gfx1250) — compile-verified
//
#include <hip/hip_runtime.h>

// RGCN: N=100000 nodes, E=1.6M edges, R=16 relations/bases, H1=64, H2=32.
#define NND 100000
#define NE  1600000
#define RR  16
#define HH1 64
#define HH2 32
#define NH1 ((size_t)NND * HH1)   // 6,400,000

typedef __attribute__((ext_vector_type(2))) float v2f;
typedef __attribute__((ext_vector_type(8))) float v8f;

// D(16x16,f32) = A(16x4,f32) x B(4x16,f32) + C  -- true-fp32 WMMA on CDNA5
static __device__ __forceinline__ v8f wmma4(v2f a, v2f b, v8f c) {
  return __builtin_amdgcn_wmma_f32_16x16x4_f32(false, a, false, b, (short)0, c,
                                               false, false);
}

// ---- counters block layout (ints): [0..15]=count [16..32]=offset
// [33..48]=cursor [49..65]=tile_offset ----

__global__ void rgcn_hist(const int* __restrict__ et, int* __restrict__ c) {
  for (int e = blockIdx.x * blockDim.x + threadIdx.x; e < NE;
       e += gridDim.x * blockDim.x)
    atomicAdd(&c[et[e]], 1);
}

__global__ void rgcn_scan(int* __restrict__ c) {
  c[16] = 0; c[49] = 0;
  for (int r = 0; r < RR; ++r) {
    c[17 + r] = c[16 + r] + c[r];          // exclusive scan -> offsets
    c[33 + r] = c[16 + r];                 // cursors
    c[50 + r] = c[49 + r] + (c[r] + 15) / 16;  // tile offsets
  }
}

__global__ void rgcn_scatter(const int* __restrict__ et, int* __restrict__ c,
                             int* __restrict__ perm) {
  for (int e = blockIdx.x * blockDim.x + threadIdx.x; e < NE;
       e += gridDim.x * blockDim.x) {
    int p = atomicAdd(&c[33 + et[e]], 1);
    perm[p] = e;
  }
}

// h = loop1[feat] + bias1
__global__ void rgcn_init_h(const int* __restrict__ feat,
                            const float* __restrict__ loop1,
                            const float* __restrict__ bias1,
                            float* __restrict__ h) {
  for (size_t i = (size_t)blockIdx.x * blockDim.x + threadIdx.x; i < NH1;
       i += (size_t)gridDim.x * blockDim.x) {
    int n = (int)(i >> 6), j = (int)(i & 63);
    h[i] = loop1[(size_t)feat[n] * HH1 + j] + bias1[j];
  }
}

// W1[r, x] = sum_b comp1[r,b] * V1[b, x]  (x = n*64+h); fully coalesced, BW-bound.
__global__ void rgcn_build_w1(const float* __restrict__ V1,
                              const float* __restrict__ comp1,
                              float* __restrict__ W1) {
  __shared__ float sc[RR * RR];
  if (threadIdx.x < RR * RR) sc[threadIdx.x] = comp1[threadIdx.x];
  __syncthreads();
  for (size_t x = (size_t)blockIdx.x * blockDim.x + threadIdx.x; x < NH1;
       x += (size_t)gridDim.x * blockDim.x) {
    float v[RR];
#pragma unroll
    for (int b = 0; b < RR; ++b) v[b] = V1[(size_t)b * NH1 + x];
#pragma unroll
    for (int r = 0; r < RR; ++r) {
      float acc = 0.f;
#pragma unroll
      for (int b = 0; b < RR; ++b) acc += sc[r * RR + b] * v[b];
      W1[(size_t)r * NH1 + x] = acc;
    }
  }
}

// layer-1 edge pass using materialized W1: 256B gather + 256B scatter-add per edge
__global__ void rgcn_edge1_w1(const int* __restrict__ src,
                              const int* __restrict__ dst,
                              const int* __restrict__ et,
                              const int* __restrict__ feat,
                              const float* __restrict__ W1,
                              float* __restrict__ h) {
  const size_t total = (size_t)NE * HH1;
  for (size_t idx = (size_t)blockIdx.x * blockDim.x + threadIdx.x; idx < total;
       idx += (size_t)gridDim.x * blockDim.x) {
    int e = (int)(idx >> 6), j = (int)(idx & 63);
    int s = feat[src[e]];
    float val = W1[(size_t)et[e] * NH1 + (size_t)s * HH1 + j];
    atomicAdd(&h[(size_t)dst[e] * HH1 + j], val);
  }
}

// fallback if workspace too small for W1: 16 gathers per edge element
__global__ void rgcn_edge1_direct(const int* __restrict__ src,
                                  const int* __restrict__ dst,
                                  const int* __restrict__ et,
                                  const int* __restrict__ feat,
                                  const float* __restrict__ V1,
                                  const float* __restrict__ comp1,
                                  float* __restrict__ h) {
  __shared__ float sc[RR * RR];
  if (threadIdx.x < RR * RR) sc[threadIdx.x] = comp1[threadIdx.x];
  __syncthreads();
  const size_t total = (size_t)NE * HH1;
  for (size_t idx = (size_t)blockIdx.x * blockDim.x + threadIdx.x; idx < total;
       idx += (size_t)gridDim.x * blockDim.x) {
    int e = (int)(idx >> 6), j = (int)(idx & 63);
    int s = feat[src[e]], r = et[e];
    float acc = 0.f;
#pragma unroll
    for (int b = 0; b < RR; ++b)
      acc += sc[r * RR + b] * V1[(size_t)b * NH1 + (size_t)s * HH1 + j];
    atomicAdd(&h[(size_t)dst[e] * HH1 + j], acc);
  }
}

__global__ void rgcn_relu(float* __restrict__ p, long long n) {
  for (long long i = (long long)blockIdx.x * blockDim.x + threadIdx.x; i < n;
       i += (long long)gridDim.x * blockDim.x)
    p[i] = fmaxf(p[i], 0.f);
}

// W2[r, y] = sum_b comp2[r,b] * V2[b, y], y in [0,2048). 16x16x16 GEMM via WMMA.
__global__ void rgcn_build_w2(const float* __restrict__ V2,
                              const float* __restrict__ comp2,
                              float* __restrict__ W2) {
  const int lane = threadIdx.x, lo = lane & 15, hi = lane >> 4;
  const int y0 = blockIdx.x * 16;  // 128 blocks of 32
  v8f c = {};
#pragma unroll
  for (int kk = 0; kk < 4; ++kk) {
    v2f a = *(const v2f*)(comp2 + lo * RR + kk * 4 + 2 * hi);  // A[M=lo][K..K+1]
    int kb = kk * 4 + 2 * hi;
    v2f b;
    b[0] = V2[(size_t)(kb + 0) * 2048 + y0 + lo];  // B[K][N=lo]
    b[1] = V2[(size_t)(kb + 1) * 2048 + y0 + lo];
    c = wmma4(a, b, c);
  }
#pragma unroll
  for (int v = 0; v < 8; ++v)
    W2[(size_t)(v + 8 * hi) * 2048 + y0 + lo] = c[v];  // D[M=v+8*hi][N=lo]
}

// out = bias2 + h @ loop2   (N x 64 @ 64 x 32), WMMA, loop2 in LDS.
__global__ void rgcn_selfloop2(const float* __restrict__ h,
                               const float* __restrict__ loop2,
                               const float* __restrict__ bias2,
                               float* __restrict__ out) {
  __shared__ float sl[HH1 * HH2];
  __shared__ float sb[HH2];
  for (int i = threadIdx.x; i < HH1 * HH2; i += blockDim.x) sl[i] = loop2[i];
  if (threadIdx.x < HH2) sb[threadIdx.x] = bias2[threadIdx.x];
  __syncthreads();
  const int lane = threadIdx.x & 31, wave = threadIdx.x >> 5;
  const int lo = lane & 15, hi = lane >> 4;
  const int ntile = (NND + 15) / 16;
  const int tile = blockIdx.x * 8 + wave;
  if (tile >= ntile) return;  // wave-uniform
  int srow = tile * 16 + lo; if (srow > NND - 1) srow = NND - 1;
  v8f c0 = {}, c1 = {};
#pragma unroll
  for (int kk = 0; kk < 16; ++kk) {
    v2f a = *(const v2f*)(h + (size_t)srow * HH1 + kk * 4 + 2 * hi);
    int kb = kk * 4 + 2 * hi;
    v2f b0, b1;
    b0[0] = sl[(kb + 0) * HH2 + lo];      b0[1] = sl[(kb + 1) * HH2 + lo];
    b1[0] = sl[(kb + 0) * HH2 + 16 + lo]; b1[1] = sl[(kb + 1) * HH2 + 16 + lo];
    c0 = wmma4(a, b0, c0);
    c1 = wmma4(a, b1, c1);
  }
#pragma unroll
  for (int v = 0; v < 8; ++v) {
    int m = tile * 16 + v + 8 * hi;
    if (m < NND) {
      out[(size_t)m * HH2 + lo]      = c0[v] + sb[lo];
      out[(size_t)m * HH2 + 16 + lo] = c1[v] + sb[16 + lo];
    }
  }
}

// relation-bucketed edge GEMM: 16 same-relation edges -> (16x64)@(64x32) WMMA,
// W2_r staged in LDS, rows scatter-added to out[dst] with f32 atomics.
__global__ void rgcn_edge2(const int* __restrict__ src,
                           const int* __restrict__ dst,
                           const int* __restrict__ perm,
                           const int* __restrict__ c,
                           const float* __restrict__ h,
                           const float* __restrict__ W2,
                           float* __restrict__ out) {
  __shared__ float sw[HH1 * HH2];
  __shared__ int ssrc[16], sdst[16];
  const int* tileoff = c + 49;
  const int* reloff  = c + 16;
  const int t = blockIdx.x;
  if (t >= tileoff[RR]) return;  // wave-uniform
  int r = RR - 1;
#pragma unroll
  for (int q = 0; q < RR; ++q)
    if (t < tileoff[q + 1]) { r = q; break; }
  const int base = reloff[r] + (t - tileoff[r]) * 16;
  const int cnt  = min(16, reloff[r + 1] - base);
  const int lane = threadIdx.x, lo = lane & 15, hi = lane >> 4;
  for (int i = lane; i < HH1 * HH2; i += 32) sw[i] = W2[(size_t)r * 2048 + i];
  if (lane < 16) {
    int ok = lane < cnt;
    int e = perm[base + (ok ? lane : 0)];
    ssrc[lane] = src[e];
    sdst[lane] = ok ? dst[e] : -1;
  }
  __syncthreads();
  const int srow = ssrc[lo];
  v8f c0 = {}, c1 = {};
#pragma unroll
  for (int kk = 0; kk < 16; ++kk) {
    v2f a = *(const v2f*)(h + (size_t)srow * HH1 + kk * 4 + 2 * hi);
    int kb = kk * 4 + 2 * hi;
    v2f b0, b1;
    b0[0] = sw[(kb + 0) * HH2 + lo];      b0[1] = sw[(kb + 1) * HH2 + lo];
    b1[0] = sw[(kb + 0) * HH2 + 16 + lo]; b1[1] = sw[(kb + 1) * HH2 + 16 + lo];
    c0 = wmma4(a, b0, c0);
    c1 = wmma4(a, b1, c1);
  }
#pragma unroll
  for (int v = 0; v < 8; ++v) {
    int d = sdst[v + 8 * hi];
    if (d >= 0) {
      atomicAdd(&out[(size_t)d * HH2 + lo],      c0[v]);
      atomicAdd(&out[(size_t)d * HH2 + 16 + lo], c1[v]);
    }
  }
}

extern "C" void kernel_launch(void* const* d_in, const int* in_sizes, int n_in,
                              void* d_out, int out_size, void* d_ws,
                              size_t ws_size, hipStream_t stream) {
  const int*   feat  = (const int*)d_in[0];
  const int*   src   = (const int*)d_in[1];
  const int*   dst   = (const int*)d_in[2];
  const int*   et    = (const int*)d_in[3];
  const float* V1    = (const float*)d_in[4];
  const float* comp1 = (const float*)d_in[5];
  const float* loop1 = (const float*)d_in[6];
  const float* bias1 = (const float*)d_in[7];
  const float* V2    = (const float*)d_in[8];
  const float* comp2 = (const float*)d_in[9];
  const float* loop2 = (const float*)d_in[10];
  const float* bias2 = (const float*)d_in[11];
  float* out = (float*)d_out;

  char* ws = (char*)d_ws;
  size_t off = 0;
  float* h = (float*)(ws + off);      off += NH1 * sizeof(float);            // 25.6MB
  float* W2ws = (float*)(ws + off);   off += (size_t)RR * 2048 * sizeof(float);
  int* perm = (int*)(ws + off);       off += (size_t)NE * sizeof(int);       // 6.4MB
  int* cnts = (int*)(ws + off);       off += 1024;
  float* W1ws = (float*)(ws + off);
  const size_t need_w1 = off + (size_t)RR * NH1 * sizeof(float);             // +410MB
  const bool use_w1 = (ws_size >= need_w1);

  hipMemsetAsync(cnts, 0, 66 * sizeof(int), stream);
  rgcn_hist<<<1024, 256, 0, stream>>>(et, cnts);
  rgcn_scan<<<1, 1, 0, stream>>>(cnts);
  rgcn_scatter<<<1024, 256, 0, stream>>>(et, cnts, perm);
  rgcn_init_h<<<4096, 256, 0, stream>>>(feat, loop1, bias1, h);
  if (use_w1) {
    rgcn_build_w1<<<8192, 256, 0, stream>>>(V1, comp1, W1ws);
    rgcn_edge1_w1<<<16384, 256, 0, stream>>>(src, dst, et, feat, W1ws, h);
  } else {
    rgcn_edge1_direct<<<16384, 256, 0, stream>>>(src, dst, et, feat, V1, comp1, h);
  }
  rgcn_relu<<<4096, 256, 0, stream>>>(h, (long long)NH1);
  rgcn_build_w2<<<128, 32, 0, stream>>>(V2, comp2, W2ws);
  rgcn_selfloop2<<<(6250 + 7) / 8, 256, 0, stream>>>(h, loop2, bias2, out);
  rgcn_edge2<<<NE / 16 + RR, 32, 0, stream>>>(src, dst, perm, cnts, h, W2ws, out);
  rgcn_relu<<<2048, 256, 0, stream>>>(out, (long long)NND * HH2);
}